// RelationNetwork_2637109919864
// MI455X (gfx1250) — compile-verified
//
#include <hip/hip_runtime.h>
#include <hip/hip_bf16.h>
#include <stdint.h>

typedef float v2f __attribute__((ext_vector_type(2)));
typedef float v8f __attribute__((ext_vector_type(8)));

#define B_    4096
#define L_    128
#define D_    100
#define S_    1024
#define K1    1224   // 2*D + S
#define N1    1024   // 2*R
#define N2    512    // R
#define EPS_  1e-5f

static __device__ __forceinline__ v8f wmma_f32(v2f a, v2f b, v8f c) {
    // V_WMMA_F32_16X16X4_F32: (neg_a, A, neg_b, B, c_mod, C, reuse_a, reuse_b)
    return __builtin_amdgcn_wmma_f32_16x16x4_f32(false, a, false, b, (short)0, c,
                                                 false, false);
}

static __device__ __forceinline__ uint32_t lds_offset(const void* p) {
    // generic pointer to LDS: low 32 bits are the wave-relative LDS byte address
    return (uint32_t)(uintptr_t)p;
}

static __device__ __forceinline__ void async_b128_to_lds(uint32_t laddr,
                                                         const float* gaddr) {
    asm volatile("global_load_async_to_lds_b128 %0, %1, off"
                 :: "v"(laddr), "v"(gaddr) : "memory");
}

static __device__ __forceinline__ void wait_async0() {
    asm volatile("s_wait_asynccnt 0x0" ::: "memory");
}

// ---------------------------------------------------------------- transpose
// in [R][C] -> out [C][R], 32x32 LDS tiles, block (32,8)
__global__ __launch_bounds__(256) void k_transpose(const float* __restrict__ in,
                                                   float* __restrict__ out,
                                                   int R, int C) {
    __shared__ float tile[32][33];
    int bx = blockIdx.x * 32, by = blockIdx.y * 32;
    int tx = threadIdx.x, ty = threadIdx.y;
    #pragma unroll
    for (int i = 0; i < 32; i += 8) {
        int r = by + ty + i, c = bx + tx;
        if (r < R && c < C) tile[ty + i][tx] = in[(long)r * C + c];
    }
    __syncthreads();
    #pragma unroll
    for (int i = 0; i < 32; i += 8) {
        int c = bx + ty + i, r = by + tx;
        if (c < C && r < R) out[(long)c * R + r] = tile[tx][ty + i];
    }
}

// ------------------------------------------------------- pool + concat -> X
__global__ __launch_bounds__(256) void k_pool(const float* __restrict__ st,
                                              const int* __restrict__ qa_ids,
                                              const int* __restrict__ qa_lens,
                                              const float* __restrict__ emb,
                                              float* __restrict__ X) {
    int b = blockIdx.x;
    int tid = threadIdx.x;
    __shared__ int ids[L_ * 2];
    ids[tid] = qa_ids[(long)b * (L_ * 2) + tid];
    __syncthreads();
    int len = qa_lens[b];
    float inv = 1.0f / (float)(len < 1 ? 1 : len);
    if (tid < 2 * D_) {
        int slot = (tid >= D_) ? 1 : 0;
        int d = tid - slot * D_;
        float acc = 0.0f;
        for (int l = 0; l < len; ++l) {
            if (l + 1 < len)
                __builtin_prefetch(emb + (long)ids[(l + 1) * 2 + slot] * D_ + d, 0, 0);
            acc += emb[(long)ids[l * 2 + slot] * D_ + d];
        }
        X[(long)b * K1 + tid] = acc * inv;
    }
    for (int j = tid; j < S_; j += 256)
        X[(long)b * K1 + 2 * D_ + j] = st[(long)b * S_ + j];
}

// ----------------------------------------------- GEMM1: Y1 = relu(X@W1 + b1)
// block tile 128M x 128N (8 waves x (16M x 128N)); B panel staged in LDS via
// async global->LDS b128, K-chunk KC=24 (51 chunks), LDS col stride 28 floats.
#define G1_KC   24
#define G1_STR  28
__global__ __launch_bounds__(256) void k_gemm1(const float* __restrict__ X,
                                               const float* __restrict__ W1T,
                                               const float* __restrict__ b1,
                                               float* __restrict__ Y1) {
    __shared__ float lds[128 * G1_STR];   // 14336 B
    int tid  = threadIdx.x;
    int lane = tid & 31;
    int wv   = tid >> 5;
    int mb   = blockIdx.x >> 3;           // 32 m-blocks
    int nb   = blockIdx.x & 7;            // 8 n-blocks
    int l15  = lane & 15;
    int kh   = (lane >> 4) * 2;
    int row  = mb * 128 + wv * 16 + l15;
    const float* xrow = X + (long)row * K1;

    // async fill mapping: c = tid>>1 (128 cols), kp = (tid&1)*12, 3 x b128
    int fc = tid >> 1;
    int fkp = (tid & 1) * 12;
    const float* gcol = W1T + (size_t)(nb * 128 + fc) * K1 + fkp;
    uint32_t lcol = lds_offset(lds) + (uint32_t)(fc * G1_STR + fkp) * 4u;

    v8f acc[8];
    #pragma unroll
    for (int j = 0; j < 8; ++j) acc[j] = (v8f){};

    for (int k0 = 0; k0 < K1; k0 += G1_KC) {
        const float* g = gcol + k0;
        async_b128_to_lds(lcol,       g);
        async_b128_to_lds(lcol + 16u, g + 4);
        async_b128_to_lds(lcol + 32u, g + 8);
        wait_async0();
        __syncthreads();
        #pragma unroll
        for (int ks = 0; ks < G1_KC / 4; ++ks) {
            int kd = ks * 4 + kh;
            v2f a = *(const v2f*)(xrow + k0 + kd);
            #pragma unroll
            for (int j = 0; j < 8; ++j) {
                v2f bf = *(const v2f*)(lds + (j * 16 + l15) * G1_STR + kd);
                acc[j] = wmma_f32(a, bf, acc[j]);
            }
        }
        __syncthreads();
    }
    // D layout: vgpr i, lanes 0-15 -> M=i ; lanes 16-31 -> M=i+8 ; N = lane&15
    int mBase = mb * 128 + wv * 16 + (lane >> 4) * 8;
    #pragma unroll
    for (int j = 0; j < 8; ++j) {
        int n = nb * 128 + j * 16 + l15;
        float bias = b1[n];
        #pragma unroll
        for (int i = 0; i < 8; ++i) {
            float y = acc[j][i] + bias;
            Y1[(long)(mBase + i) * N1 + n] = y > 0.0f ? y : 0.0f;
        }
    }
}

// ---------------------------------------- column batch-stats -> scale/shift
__global__ __launch_bounds__(256) void k_stats(const float* __restrict__ in,
                                               const float* __restrict__ g,
                                               const float* __restrict__ beta,
                                               float* __restrict__ s,
                                               float* __restrict__ t, int N) {
    __shared__ float ssum[256], ssq[256];
    int tid  = threadIdx.x;
    int col  = blockIdx.x * 32 + (tid & 31);
    int part = tid >> 5;
    const float* p = in + (long)part * 512 * N + col;
    float sum = 0.0f, sq = 0.0f;
    for (int r = 0; r < 512; ++r) {
        float v = p[(long)r * N];
        sum += v; sq += v * v;
    }
    ssum[tid] = sum; ssq[tid] = sq;
    __syncthreads();
    if (part == 0) {
        #pragma unroll
        for (int q = 1; q < 8; ++q) { sum += ssum[tid + 32 * q]; sq += ssq[tid + 32 * q]; }
        float mean = sum * (1.0f / (float)B_);
        float var  = sq  * (1.0f / (float)B_) - mean * mean;
        float sc   = g[col] * rsqrtf(var + EPS_);
        s[col] = sc;
        t[col] = beta[col] - mean * sc;
    }
}

// ------------------------- GEMM2: Z = (BN1(Y1)) @ W2 + b2  (BN1 fused in A)
// block tile 128M x 128N, KC=32 (32 chunks), LDS col stride 36 floats.
#define G2_KC   32
#define G2_STR  36
__global__ __launch_bounds__(256) void k_gemm2(const float* __restrict__ Y1,
                                               const float* __restrict__ s1,
                                               const float* __restrict__ t1,
                                               const float* __restrict__ W2T,
                                               const float* __restrict__ b2,
                                               float* __restrict__ Z) {
    __shared__ float lds[128 * G2_STR];   // 18432 B
    int tid  = threadIdx.x;
    int lane = tid & 31;
    int wv   = tid >> 5;
    int mb   = blockIdx.x >> 2;           // 32 m-blocks
    int nb   = blockIdx.x & 3;            // 4 n-blocks
    int l15  = lane & 15;
    int kh   = (lane >> 4) * 2;
    int row  = mb * 128 + wv * 16 + l15;
    const float* yrow = Y1 + (long)row * N1;

    // async fill mapping: c = tid>>1, kp = (tid&1)*16, 4 x b128
    int fc = tid >> 1;
    int fkp = (tid & 1) * 16;
    const float* gcol = W2T + (size_t)(nb * 128 + fc) * N1 + fkp;
    uint32_t lcol = lds_offset(lds) + (uint32_t)(fc * G2_STR + fkp) * 4u;

    v8f acc[8];
    #pragma unroll
    for (int j = 0; j < 8; ++j) acc[j] = (v8f){};

    for (int k0 = 0; k0 < N1; k0 += G2_KC) {
        const float* g = gcol + k0;
        async_b128_to_lds(lcol,       g);
        async_b128_to_lds(lcol + 16u, g + 4);
        async_b128_to_lds(lcol + 32u, g + 8);
        async_b128_to_lds(lcol + 48u, g + 12);
        wait_async0();
        __syncthreads();
        #pragma unroll
        for (int ks = 0; ks < G2_KC / 4; ++ks) {
            int kd = ks * 4 + kh;
            int kk = k0 + kd;
            v2f y  = *(const v2f*)(yrow + kk);
            v2f sc = *(const v2f*)(s1 + kk);
            v2f sh = *(const v2f*)(t1 + kk);
            v2f a  = y * sc + sh;
            #pragma unroll
            for (int j = 0; j < 8; ++j) {
                v2f bf = *(const v2f*)(lds + (j * 16 + l15) * G2_STR + kd);
                acc[j] = wmma_f32(a, bf, acc[j]);
            }
        }
        __syncthreads();
    }
    int mBase = mb * 128 + wv * 16 + (lane >> 4) * 8;
    #pragma unroll
    for (int j = 0; j < 8; ++j) {
        int n = nb * 128 + j * 16 + l15;
        float bias = b2[n];
        #pragma unroll
        for (int i = 0; i < 8; ++i)
            Z[(long)(mBase + i) * N2 + n] = acc[j][i] + bias;
    }
}

// ------------------- final: out[m] = relu(BN2(Z[m])) . W3 + b3 (wave/row)
__global__ __launch_bounds__(256) void k_final(const float* __restrict__ Z,
                                               const float* __restrict__ s2,
                                               const float* __restrict__ t2,
                                               const float* __restrict__ W3,
                                               const float* __restrict__ b3,
                                               float* __restrict__ out) {
    int wid  = (blockIdx.x * 256 + threadIdx.x) >> 5;
    int lane = threadIdx.x & 31;
    const float* zrow = Z + (long)wid * N2;
    float acc = 0.0f;
    for (int k = lane; k < N2; k += 32) {
        float v = zrow[k] * s2[k] + t2[k];
        v = v > 0.0f ? v : 0.0f;
        acc += v * W3[k];
    }
    #pragma unroll
    for (int off = 16; off > 0; off >>= 1) acc += __shfl_down(acc, off, 32);
    if (lane == 0) out[wid] = acc + b3[0];
}

extern "C" void kernel_launch(void* const* d_in, const int* in_sizes, int n_in,
                              void* d_out, int out_size, void* d_ws, size_t ws_size,
                              hipStream_t stream) {
    const float* st      = (const float*)d_in[0];
    const int*   qa_ids  = (const int*)  d_in[1];
    const int*   qa_lens = (const int*)  d_in[2];
    const float* emb     = (const float*)d_in[3];
    const float* W1      = (const float*)d_in[4];
    const float* b1      = (const float*)d_in[5];
    const float* g1      = (const float*)d_in[6];
    const float* beta1   = (const float*)d_in[7];
    const float* W2      = (const float*)d_in[8];
    const float* b2      = (const float*)d_in[9];
    const float* g2      = (const float*)d_in[10];
    const float* beta2   = (const float*)d_in[11];
    const float* W3      = (const float*)d_in[12];
    const float* b3      = (const float*)d_in[13];
    float* out = (float*)d_out;

    float* ws = (float*)d_ws;
    // workspace layout (floats); Z aliases X (X dead after GEMM1)
    const size_t o_X   = 0;                       // 4096*1224 = 5,013,504
    const size_t o_Y1  = o_X  + (size_t)B_ * K1;  // 4096*1024 = 4,194,304
    const size_t o_W1T = o_Y1 + (size_t)B_ * N1;  // 1024*1224 = 1,253,376
    const size_t o_W2T = o_W1T + (size_t)N1 * K1; // 512*1024  =   524,288
    const size_t o_s1  = o_W2T + (size_t)N2 * N1; // 1024
    const size_t o_t1  = o_s1 + N1;               // 1024
    const size_t o_s2  = o_t1 + N1;               // 512
    const size_t o_t2  = o_s2 + N2;               // 512
    float* X   = ws + o_X;
    float* Z   = ws + o_X;   // alias
    float* Y1  = ws + o_Y1;
    float* W1T = ws + o_W1T;
    float* W2T = ws + o_W2T;
    float* s1  = ws + o_s1;
    float* t1  = ws + o_t1;
    float* s2  = ws + o_s2;
    float* t2  = ws + o_t2;

    dim3 tb(32, 8, 1);
    // W1 [K1][N1] -> W1T [N1][K1]
    k_transpose<<<dim3((N1 + 31) / 32, (K1 + 31) / 32, 1), tb, 0, stream>>>(W1, W1T, K1, N1);
    // W2 [N1][N2] -> W2T [N2][N1]
    k_transpose<<<dim3((N2 + 31) / 32, (N1 + 31) / 32, 1), tb, 0, stream>>>(W2, W2T, N1, N2);

    k_pool<<<B_, 256, 0, stream>>>(st, qa_ids, qa_lens, emb, X);

    // 32 m-blocks * 8 n-blocks = 256 blocks of 8 waves
    k_gemm1<<<256, 256, 0, stream>>>(X, W1T, b1, Y1);
    k_stats<<<N1 / 32, 256, 0, stream>>>(Y1, g1, beta1, s1, t1, N1);

    // 32 m-blocks * 4 n-blocks = 128 blocks
    k_gemm2<<<128, 256, 0, stream>>>(Y1, s1, t1, W2T, b2, Z);
    k_stats<<<N2 / 32, 256, 0, stream>>>(Z, g2, beta2, s2, t2, N2);

    // 4096 waves -> 512 blocks
    k_final<<<512, 256, 0, stream>>>(Z, s2, t2, W3, b3, out);
}